// BiRNNLM_41850161333039
// MI455X (gfx1250) — compile-verified
//
#include <hip/hip_runtime.h>
#include <hip/hip_bf16.h>
#include <math.h>

typedef __attribute__((ext_vector_type(2))) float v2f;
typedef __attribute__((ext_vector_type(8))) float v8f;

#define SEQ   128
#define BATCH 32
#define EMB   32
#define HID   8
#define VOC   32000
#define ROWS  (SEQ*BATCH)   /* 4096 */
#define KDIM  16            /* 2*HID */

// ---------------------------------------------------------------------------
// Kernel 1: embedding lookup + forward/backward RNN scans.
// One wave (32 threads) per (batch, direction); 64 blocks. Writes the state
// table Hcat[(t*B+b)*16 + dir*8 + j] = h BEFORE the update at step t
// (matches jax.lax.scan returning the carry, not h_new).
// ---------------------------------------------------------------------------
__global__ __launch_bounds__(32) void birnn_scan_kernel(
    const int*   __restrict__ idx,     // (S,B)
    const float* __restrict__ lut,     // (V,E)
    const float* __restrict__ wxf,     // (E,H)
    const float* __restrict__ whf,     // (H,H)
    const float* __restrict__ wxb,     // (E,H)
    const float* __restrict__ whb,     // (H,H)
    const float* __restrict__ hf0,     // (H)
    const float* __restrict__ hb0,     // (H)
    const float* __restrict__ bx,      // (H)
    const float* __restrict__ bhf,     // (H)
    const float* __restrict__ bhb,     // (H)
    float* __restrict__ Hcat)          // (S*B, 16)
{
    const int b   = blockIdx.x & (BATCH - 1);
    const int dir = blockIdx.x >> 5;          // 0 = forward, 1 = backward
    const int tid = threadIdx.x;              // 0..31

    __shared__ float xs[EMB];
    __shared__ float hs[HID];

    const float* Wx = dir ? wxb : wxf;
    const float* Wh = dir ? whb : whf;
    const float* bh = dir ? bhb : bhf;
    const float* h0 = dir ? hb0 : hf0;

    if (tid < HID) hs[tid] = h0[tid];
    __syncthreads();

    for (int step = 0; step < SEQ; ++step) {
        const int t = dir ? (SEQ - 1 - step) : step;
        const int token = idx[t * BATCH + b];
        xs[tid] = lut[(size_t)token * EMB + tid];
        __syncthreads();

        float hnew = 0.f;
        if (tid < HID) {
            // table stores the carry BEFORE consuming x[t]
            Hcat[(size_t)(t * BATCH + b) * KDIM + dir * HID + tid] = hs[tid];
            float acc = bx[tid] + bh[tid];
#pragma unroll
            for (int e = 0; e < EMB; ++e) acc += xs[e] * Wx[e * HID + tid];
#pragma unroll
            for (int k = 0; k < HID; ++k) acc += hs[k] * Wh[k * HID + tid];
            hnew = tanhf(acc);
        }
        __syncthreads();
        if (tid < HID) hs[tid] = hnew;
        __syncthreads();
    }
}

// ---------------------------------------------------------------------------
// Kernel 2: logits = Hcat @ Wo + bias_o, then log_softmax over V.
// Two-pass recompute: pass 1 keeps only online (max, sumexp) per row;
// pass 2 recomputes logits and writes normalized values (NT stores).
// WMMA f32 16x16x4, K=16 via 4 chained WMMAs.
// Block: 256 threads = 8 waves, each wave a 16-wide N slice; 250 iterations
// cover N=32000. One block per 16 M rows -> 256 blocks.
// ---------------------------------------------------------------------------
__global__ __launch_bounds__(256) void logits_logsoftmax_kernel(
    const float* __restrict__ Hcat,   // (4096, 16)
    const float* __restrict__ Wo,     // (16, V)
    const float* __restrict__ bo,     // (V)
    float* __restrict__ out)          // (4096, V)
{
    const int wave = threadIdx.x >> 5;   // 0..7
    const int lane = threadIdx.x & 31;
    const int half = lane >> 4;          // 0 or 1
    const int l16  = lane & 15;
    const int mt   = blockIdx.x;         // 0..255

    __shared__ float red_m[8][16];
    __shared__ float red_s[8][16];
    __shared__ float fin[16];

    // ---- load the 16x16 A tile once, split into 4 K-slices of 4 ----
    // A 16x4 f32 layout: VGPR0 = K=2*half, VGPR1 = K=1+2*half, M = lane%16
    v2f a[4];
    const int arow = mt * 16 + l16;
#pragma unroll
    for (int kk = 0; kk < 4; ++kk) {
        a[kk].x = Hcat[(size_t)arow * KDIM + kk * 4 + 2 * half];
        a[kk].y = Hcat[(size_t)arow * KDIM + kk * 4 + 1 + 2 * half];
    }

    float rm[8], rs[8];
#pragma unroll
    for (int i = 0; i < 8; ++i) { rm[i] = -INFINITY; rs[i] = 0.f; }

    const int NITER = VOC / 128;   // 250

    // ---------------- pass 1: online max / sum-exp ----------------
    for (int it = 0; it < NITER; ++it) {
        const int col = it * 128 + wave * 16 + l16;
        v8f c = {};
#pragma unroll
        for (int kk = 0; kk < 4; ++kk) {
            v2f b;
            b.x = Wo[(size_t)(kk * 4 + 2 * half) * VOC + col];
            b.y = Wo[(size_t)(kk * 4 + 1 + 2 * half) * VOC + col];
            c = __builtin_amdgcn_wmma_f32_16x16x4_f32(false, a[kk], false, b,
                                                      (short)0, c, false, false);
        }
        const float bias = bo[col];
#pragma unroll
        for (int i = 0; i < 8; ++i) {
            float v  = c[i] + bias;
            float nm = fmaxf(rm[i], v);
            rs[i] = rs[i] * __expf(rm[i] - nm) + __expf(v - nm);
            rm[i] = nm;
        }
    }

    // intra-wave reduction across the 16 lanes of each half
#pragma unroll
    for (int off = 1; off < 16; off <<= 1) {
#pragma unroll
        for (int i = 0; i < 8; ++i) {
            float om = __shfl_xor(rm[i], off, 32);
            float os = __shfl_xor(rs[i], off, 32);
            float nm = fmaxf(rm[i], om);
            rs[i] = rs[i] * __expf(rm[i] - nm) + os * __expf(om - nm);
            rm[i] = nm;
        }
    }
    if (l16 == 0) {
#pragma unroll
        for (int i = 0; i < 8; ++i) {
            red_m[wave][i + 8 * half] = rm[i];
            red_s[wave][i + 8 * half] = rs[i];
        }
    }
    __syncthreads();

    // cross-wave combine: thread r (<16) owns row r
    if (threadIdx.x < 16) {
        float M = -INFINITY, SS = 0.f;
        for (int w = 0; w < 8; ++w) {
            float om = red_m[w][threadIdx.x];
            float os = red_s[w][threadIdx.x];
            float nm = fmaxf(M, om);
            SS = SS * __expf(M - nm) + os * __expf(om - nm);
            M = nm;
        }
        fin[threadIdx.x] = M + logf(SS);   // logsumexp per row
    }
    __syncthreads();

    float sub[8];
#pragma unroll
    for (int i = 0; i < 8; ++i) sub[i] = fin[i + 8 * half];

    // ---------------- pass 2: recompute + write normalized ----------------
    for (int it = 0; it < NITER; ++it) {
        const int col = it * 128 + wave * 16 + l16;
        v8f c = {};
#pragma unroll
        for (int kk = 0; kk < 4; ++kk) {
            v2f b;
            b.x = Wo[(size_t)(kk * 4 + 2 * half) * VOC + col];
            b.y = Wo[(size_t)(kk * 4 + 1 + 2 * half) * VOC + col];
            c = __builtin_amdgcn_wmma_f32_16x16x4_f32(false, a[kk], false, b,
                                                      (short)0, c, false, false);
        }
        const float bias = bo[col];
#pragma unroll
        for (int i = 0; i < 8; ++i) {
            float v = c[i] + bias - sub[i];
            __builtin_nontemporal_store(
                v, &out[(size_t)(mt * 16 + i + 8 * half) * VOC + col]);
        }
    }
}

extern "C" void kernel_launch(void* const* d_in, const int* in_sizes, int n_in,
                              void* d_out, int out_size, void* d_ws, size_t ws_size,
                              hipStream_t stream) {
    const int*   idx = (const int*)  d_in[0];   // input_batch (S,B) int32
    const float* lut = (const float*)d_in[1];   // lookup (V,E)
    const float* wxf = (const float*)d_in[2];   // (E,H)
    const float* whf = (const float*)d_in[3];   // (H,H)
    const float* wxb = (const float*)d_in[4];   // (E,H)
    const float* whb = (const float*)d_in[5];   // (H,H)
    const float* wo  = (const float*)d_in[6];   // (2H,V)
    const float* hf0 = (const float*)d_in[7];   // (H)
    const float* hb0 = (const float*)d_in[8];   // (H)
    const float* bx  = (const float*)d_in[9];   // (H)
    const float* bhf = (const float*)d_in[10];  // (H)
    const float* bhb = (const float*)d_in[11];  // (H)
    const float* bo  = (const float*)d_in[12];  // (V)
    float* out  = (float*)d_out;
    float* Hcat = (float*)d_ws;                 // 4096*16 floats = 256 KB

    birnn_scan_kernel<<<64, 32, 0, stream>>>(idx, lut, wxf, whf, wxb, whb,
                                             hf0, hb0, bx, bhf, bhb, Hcat);
    logits_logsoftmax_kernel<<<ROWS / 16, 256, 0, stream>>>(Hcat, wo, bo, out);
}